// Features_Loss_24507083391416
// MI455X (gfx1250) — compile-verified
//
#include <hip/hip_runtime.h>

typedef __attribute__((ext_vector_type(2))) float v2f;
typedef __attribute__((ext_vector_type(8))) float v8f;

#define PNUM 16
#define DDIM 2048

// One wave (32 lanes) computes one group's 16x16 Gram matrix via
// V_WMMA_F32_16X16X4_F32, K-loop over D=2048 in steps of 4.
// A(16x4) and B(4x16)=A^T have identical per-lane VGPR layouts for this tile,
// so a single b64 load per lane feeds both WMMA operands.
__global__ __launch_bounds__(256) void gram_upper_kernel(
    const float* __restrict__ f, float* __restrict__ ws, int groups) {
  const int wave = (int)((blockIdx.x * blockDim.x + threadIdx.x) >> 5);
  const int lane = (int)(threadIdx.x & 31);
  if (wave >= groups) return;  // wave-uniform branch: EXEC stays all-ones

  const int row    = lane & 15;        // M (and N) index this lane owns
  const int colOff = (lane >> 4) << 1; // lanes 16..31 handle K+2,K+3
  const float* base =
      f + (size_t)wave * (PNUM * DDIM) + (size_t)row * DDIM + colOff;

  v8f c0 = {0.f, 0.f, 0.f, 0.f, 0.f, 0.f, 0.f, 0.f};
  v8f c1 = {0.f, 0.f, 0.f, 0.f, 0.f, 0.f, 0.f, 0.f};
  float sq = 0.f;  // per-lane partial of trace = sum_i ||f_i||^2

  for (int k = 0; k < DDIM; k += 32) {
    v2f a[8];
#pragma unroll
    for (int u = 0; u < 8; ++u)
      a[u] = *(const v2f*)(base + k + 4 * u);  // global_load_b64, 8 in flight
#pragma unroll
    for (int u = 0; u < 8; ++u) {
      sq += a[u].x * a[u].x + a[u].y * a[u].y;  // co-executes with WMMA pipe
      if (u & 1)
        c1 = __builtin_amdgcn_wmma_f32_16x16x4_f32(
            false, a[u], false, a[u], (short)0, c1, false, false);
      else
        c0 = __builtin_amdgcn_wmma_f32_16x16x4_f32(
            false, a[u], false, a[u], (short)0, c0, false, false);
    }
  }

  // Sum of all 256 Gram entries: each lane holds 8 (two half-accumulators).
  float s_all = (c0[0] + c1[0]) + (c0[1] + c1[1]) + (c0[2] + c1[2]) +
                (c0[3] + c1[3]) + (c0[4] + c1[4]) + (c0[5] + c1[5]) +
                (c0[6] + c1[6]) + (c0[7] + c1[7]);
  float s_diag = sq;

  // wave32 butterfly reduction
#pragma unroll
  for (int off = 16; off >= 1; off >>= 1) {
    s_all  += __shfl_xor(s_all, off, 32);
    s_diag += __shfl_xor(s_diag, off, 32);
  }

  if (lane == 0) ws[wave] = 0.5f * (s_all - s_diag);  // sum_{i<j} gram_ij
}

// Deterministic tree reduction of per-group sums -> final scalar.
__global__ __launch_bounds__(256) void finalize_kernel(
    const float* __restrict__ ws, float* __restrict__ out, int groups) {
  __shared__ float sm[256];
  float s = 0.f;
  for (int i = (int)threadIdx.x; i < groups; i += 256) s += ws[i];
  sm[threadIdx.x] = s;
  __syncthreads();
#pragma unroll
  for (int stride = 128; stride > 0; stride >>= 1) {
    if ((int)threadIdx.x < stride) sm[threadIdx.x] += sm[threadIdx.x + stride];
    __syncthreads();
  }
  if (threadIdx.x == 0) {
    const float pairs = (float)(PNUM * (PNUM - 1)) * 0.5f;  // 120
    // mean_g[(pairs - upper_g)/pairs] = 1 - (sum_g upper_g)/(G*pairs)
    out[0] = 1.0f - sm[0] / ((float)groups * pairs);
  }
}

extern "C" void kernel_launch(void* const* d_in, const int* in_sizes, int n_in,
                              void* d_out, int out_size, void* d_ws,
                              size_t ws_size, hipStream_t stream) {
  const float* features = (const float*)d_in[0];
  float* out = (float*)d_out;
  float* ws  = (float*)d_ws;

  const int groups = in_sizes[0] / (PNUM * DDIM);  // 16384*2048 / (16*2048) = 1024
  const int threads = 256;                          // 8 waves per block
  const int blocks  = (groups * 32 + threads - 1) / threads;

  gram_upper_kernel<<<blocks, threads, 0, stream>>>(features, ws, groups);
  finalize_kernel<<<1, threads, 0, stream>>>(ws, out, groups);
}